// LaplacianSpatioTemporal_6055903888131
// MI455X (gfx1250) — compile-verified
//
#include <hip/hip_runtime.h>
#include <cstdint>

#ifndef __has_builtin
#define __has_builtin(x) 0
#endif

// gfx1250 async-copy builtin takes (global int4*, lds int4*, imm offset, imm cpol).
// clang prints AS1 as '__device__' in HIP mode — confirmed by round-1 diagnostic.
typedef int v4i __attribute__((vector_size(4 * sizeof(int))));
#define GLOBAL_V4I(p) ((__attribute__((address_space(1))) v4i*)(p))
#define LDS_V4I(p)    ((__attribute__((address_space(3))) v4i*)(p))

#if __has_builtin(__builtin_amdgcn_global_load_async_to_lds_b128)
#define USE_ASYNC_LDS 1
#else
#define USE_ASYNC_LDS 0
#endif

static constexpr int kC = 64, kT = 16, kH = 128, kW = 128;
static constexpr int kPlane = kH * kW;                          // 16384 f32 = 64 KiB
static constexpr int kThreads = 256;                            // 8 waves (wave32)
static constexpr int kVec4PerThread = kPlane / (kThreads * 4);  // 16

__device__ __forceinline__ float softplus_f(float x) {
    // numerically safe softplus; inputs are ~0.1 so this is exact enough
    return fmaxf(x, 0.0f) + log1pf(expf(-fabsf(x)));
}

__global__ __launch_bounds__(kThreads)
void lap_spatiotemporal_kernel(const float* __restrict__ u,
                               const float* __restrict__ Ds,
                               const float* __restrict__ Dt,
                               float* __restrict__ out)
{
    __shared__ float tile[kPlane];   // one (h,w) plane, 64 KiB of the 320 KB WGP LDS

    const int plane_id = blockIdx.x;            // ((b*C + c)*T + t)
    const int t   = plane_id % kT;
    const int c   = (plane_id / kT) % kC;
    const int tid = threadIdx.x;

    const float* __restrict__ uplane = u + (size_t)plane_id * kPlane;

    // ---- Stage this plane into LDS (CDNA5 async copy path, ASYNCcnt-tracked) ----
#if USE_ASYNC_LDS
#pragma unroll
    for (int k = 0; k < kVec4PerThread; ++k) {
        const int f = (tid + k * kThreads) * 4;   // float index; 16B per lane per issue
        __builtin_amdgcn_global_load_async_to_lds_b128(
            GLOBAL_V4I(uplane + f), LDS_V4I(&tile[f]), 0, 0);
    }
#if __has_builtin(__builtin_amdgcn_s_wait_asynccnt)
    __builtin_amdgcn_s_wait_asynccnt(0);
#else
    asm volatile("s_wait_asynccnt 0x0" ::: "memory");
#endif
#else
#pragma unroll
    for (int k = 0; k < kVec4PerThread; ++k) {
        const int f = (tid + k * kThreads) * 4;
        *(float4*)(&tile[f]) = *(const float4*)(uplane + f);
    }
#endif
    __syncthreads();

    // ---- Per-channel softplus coefficients (uniform per block) ----
    const float cs0 = softplus_f(Ds[0 * kC + c]);   // d = 1
    const float cs1 = softplus_f(Ds[1 * kC + c]);   // d = 4
    const float cs2 = softplus_f(Ds[2 * kC + c]);   // d = 16
    const float ct0 = softplus_f(Dt[0 * kC + c]);   // dt = 1
    const float ct1 = softplus_f(Dt[1 * kC + c]);   // dt = 2

    // Temporal neighbor planes (clamped at t=0); center values only -> L2-served.
    const int t1 = (t >= 1) ? (t - 1) : 0;
    const int t2 = (t >= 2) ? (t - 2) : 0;
    const float* __restrict__ up1 = uplane + (ptrdiff_t)(t1 - t) * kPlane;
    const float* __restrict__ up2 = uplane + (ptrdiff_t)(t2 - t) * kPlane;
    float* __restrict__ oplane = out + (size_t)plane_id * kPlane;

    for (int k = 0; k < kVec4PerThread; ++k) {
        const int v4   = tid + k * kThreads;       // float4 slot in plane
        const int base = v4 * 4;                   // first float of the slot
        const int h    = base >> 7;                // base / kW
        const int w0   = base & (kW - 1);          // 4-aligned, so all 4 in one row

        const float4 p1 = *(const float4*)(up1 + base);
        const float4 p2 = *(const float4*)(up2 + base);
        const float* p1v = &p1.x;
        const float* p2v = &p2.x;
        float4 o;
        float* ov = &o.x;

#pragma unroll
        for (int j = 0; j < 4; ++j) {
            const int w = w0 + j;
            const float uc = tile[h * kW + w];
            float acc = 0.0f;
#pragma unroll
            for (int i = 0; i < 3; ++i) {
                const int   d  = (i == 0) ? 1 : (i == 1) ? 4 : 16;
                const float cs = (i == 0) ? cs0 : (i == 1) ? cs1 : cs2;
                const int hm = (h - d > 0) ? (h - d) : 0;            // replicate pad
                const int hp = (h + d < kH - 1) ? (h + d) : (kH - 1);
                const int wm = (w - d > 0) ? (w - d) : 0;
                const int wp = (w + d < kW - 1) ? (w + d) : (kW - 1);
                const float lap = tile[hm * kW + w] + tile[hp * kW + w]
                                + tile[h * kW + wm] + tile[h * kW + wp]
                                - 4.0f * uc;
                acc = fmaf(cs, lap, acc);
            }
            acc = fmaf(ct0, p1v[j] - uc, acc);     // u[t-1] - u
            acc = fmaf(ct1, p2v[j] - uc, acc);     // u[t-2] - u
            ov[j] = acc;
        }
        *(float4*)(oplane + base) = o;
    }
}

extern "C" void kernel_launch(void* const* d_in, const int* in_sizes, int n_in,
                              void* d_out, int out_size, void* d_ws, size_t ws_size,
                              hipStream_t stream) {
    (void)n_in; (void)out_size; (void)d_ws; (void)ws_size;
    const float* u  = (const float*)d_in[0];   // (B, C, T, H, W) f32
    const float* Ds = (const float*)d_in[1];   // (3, C) f32
    const float* Dt = (const float*)d_in[2];   // (2, C) f32
    float* out = (float*)d_out;

    const int planes = in_sizes[0] / kPlane;   // B*C*T = 4096
    lap_spatiotemporal_kernel<<<planes, kThreads, 0, stream>>>(u, Ds, Dt, out);
}